// Model_73710228734964
// MI455X (gfx1250) — compile-verified
//
#include <hip/hip_runtime.h>
#include <hip/hip_bf16.h>
#include <math.h>

// ---------------------------------------------------------------------------
// GCN pipeline for MI455X (gfx1250, wave32).
//   deg -> dinv -> embed -> 8x { bn_stats, BN+WMMA(8x8), zero agg, edge
//   scatter (L2-resident atomics), residual+relu } -> pool -> WMMA MLP.
// Dense transforms use V_WMMA_F32_16X16X4_F32 (exact fp32, K=4, two per K=8).
// Working set (64MB edges + 3x16MB node arrays + 2MB dinv) fits MI455X's
// 192MB L2, so default RT temporal hints are correct; scatter is L2-bound.
// ---------------------------------------------------------------------------

typedef __attribute__((ext_vector_type(2))) float v2f;
typedef __attribute__((ext_vector_type(8))) float v8f;

#define CHN 8
#define HID 128
#define BN_EPS 1e-5f

// ---------------- small utility kernels ----------------

__global__ void k_fill(float* __restrict__ p, float v, int n) {
  int i = blockIdx.x * blockDim.x + threadIdx.x;
  if (i < n) p[i] = v;
}

__global__ void k_deg_acc(const int* __restrict__ dst, float* __restrict__ deg, int e) {
  int i = blockIdx.x * blockDim.x + threadIdx.x;
  if (i < e) atomicAdd(&deg[dst[i]], 1.0f);
}

__global__ void k_dinv(float* __restrict__ deg, int n) {
  int i = blockIdx.x * blockDim.x + threadIdx.x;
  if (i < n) deg[i] = rsqrtf(deg[i]);   // deg >= 1 due to self loops
}

__global__ void k_embed(const int* __restrict__ x, const float* __restrict__ emb,
                        float* __restrict__ h, int total) {
  int t = blockIdx.x * blockDim.x + threadIdx.x;
  if (t < total) h[t] = emb[x[t >> 3] * CHN + (t & 7)];
}

// ---------------- batch-norm statistics ----------------
// stats[0..7] = per-channel sum, stats[8..15] = per-channel sum of squares.
// Grid-stride is a multiple of 8 so each thread touches a single channel.
__global__ __launch_bounds__(256) void k_bn_stats(const float* __restrict__ h,
                                                  float* __restrict__ stats, int total) {
  __shared__ float ssum[CHN];
  __shared__ float ssq[CHN];
  if (threadIdx.x < CHN) { ssum[threadIdx.x] = 0.f; ssq[threadIdx.x] = 0.f; }
  __syncthreads();
  int t0 = blockIdx.x * blockDim.x + threadIdx.x;
  int stride = gridDim.x * blockDim.x;          // multiple of 8
  float s = 0.f, s2 = 0.f;
  for (int t = t0; t < total; t += stride) {
    float v = h[t];
    s += v; s2 += v * v;
  }
  int c = t0 & 7;                               // fixed channel per thread
  atomicAdd(&ssum[c], s);
  atomicAdd(&ssq[c], s2);
  __syncthreads();
  if (threadIdx.x < CHN) {
    atomicAdd(&stats[threadIdx.x], ssum[threadIdx.x]);
    atomicAdd(&stats[CHN + threadIdx.x], ssq[threadIdx.x]);
  }
}

// ---------------- BN + 8x8 linear via fp32 WMMA ----------------
// One wave handles 16 nodes: A = 16x8 normalized activations (two 16x4
// slices), B = conv_w (8x8, N padded to 16 with zeros), D = 16x16 fp32.
// fp32 A layout (ISA 7.12.2): lane<16 holds K={0,1}, lane>=16 holds K={2,3}.
__global__ __launch_bounds__(256) void k_transform(
    const float* __restrict__ h, const float* __restrict__ stats,
    const float* __restrict__ gamma, const float* __restrict__ beta,
    const float* __restrict__ W, float* __restrict__ hw, int n) {
  const int lane = threadIdx.x & 31;
  const int wave = blockIdx.x * (blockDim.x >> 5) + (threadIdx.x >> 5);
  const int base = wave * 16;
  if (base >= n) return;                         // wave-uniform exit (EXEC all-1)
  const bool hiHalf = lane >= 16;
  const int m = lane & 15;
  const int row = min(base + m, n - 1);          // clamp tail loads
  const float invN = 1.0f / (float)n;

  const int c0 = hiHalf ? 2 : 0;
  v2f A0, A1;
#pragma unroll
  for (int q = 0; q < 2; ++q) {
    {
      int c = c0 + q;                            // K = 0..3 slice
      float mu = stats[c] * invN;
      float var = stats[CHN + c] * invN - mu * mu;
      float sc = rsqrtf(var + BN_EPS) * gamma[c];
      float sh = beta[c] - mu * sc;
      A0[q] = h[(size_t)row * CHN + c] * sc + sh;
    }
    {
      int c = c0 + 4 + q;                        // K = 4..7 slice
      float mu = stats[c] * invN;
      float var = stats[CHN + c] * invN - mu * mu;
      float sc = rsqrtf(var + BN_EPS) * gamma[c];
      float sh = beta[c] - mu * sc;
      A1[q] = h[(size_t)row * CHN + c] * sc + sh;
    }
  }

  // B: 4x16 slices of conv_w (row = input channel K, col = output channel N).
  const int nn = m;
  const int k0 = hiHalf ? 2 : 0;
  const bool cvalid = nn < CHN;
  v2f B0, B1;
  B0[0] = cvalid ? W[(k0    ) * CHN + nn] : 0.f;
  B0[1] = cvalid ? W[(k0 + 1) * CHN + nn] : 0.f;
  B1[0] = cvalid ? W[(k0 + 4) * CHN + nn] : 0.f;
  B1[1] = cvalid ? W[(k0 + 5) * CHN + nn] : 0.f;

  v8f acc = {};
  acc = __builtin_amdgcn_wmma_f32_16x16x4_f32(false, A0, false, B0, (short)0, acc, false, false);
  acc = __builtin_amdgcn_wmma_f32_16x16x4_f32(false, A1, false, B1, (short)0, acc, false, false);

  // D: VGPR j -> (M=j, N=lane) for lanes 0-15, (M=j+8, N=lane-16) for 16-31.
  // Fast path for full tiles: one base address per lane, constant byte
  // offsets fold into the global_store_b32 ioffset field (no per-row guards).
  const int outRow0 = base + (hiHalf ? 8 : 0);
  float* __restrict__ outp = hw + (size_t)outRow0 * CHN + nn;
  if (base + 16 <= n) {                          // wave-uniform fast path
    if (cvalid) {
#pragma unroll
      for (int j = 0; j < 8; ++j) outp[j * CHN] = acc[j];
    }
  } else {                                       // at most one tail wave
    if (cvalid) {
#pragma unroll
      for (int j = 0; j < 8; ++j)
        if (outRow0 + j < n) outp[j * CHN] = acc[j];
    }
  }
}

// ---------------- normalized edge scatter-add ----------------
// One thread per edge (plus N self loops).  hw/agg/dinv (~34 MB) plus the
// 64 MB edge stream are L2-resident on MI455X's 192 MB L2.
__global__ __launch_bounds__(256) void k_scatter(
    const int* __restrict__ src, const int* __restrict__ dst,
    const float* __restrict__ dinv, const float* __restrict__ hw,
    float* __restrict__ agg, int nE, int nN) {
  int t = blockIdx.x * blockDim.x + threadIdx.x;
  int total = nE + nN;
  if (t >= total) return;
  int s, d;
  if (t < nE) { s = src[t]; d = dst[t]; }
  else        { s = t - nE; d = s; }             // self loop
  float w = dinv[s] * dinv[d];
  const float4* p = (const float4*)(hw + (size_t)s * CHN);
  float4 lo = p[0], hi = p[1];
  float* ap = agg + (size_t)d * CHN;
  atomicAdd(ap + 0, w * lo.x);
  atomicAdd(ap + 1, w * lo.y);
  atomicAdd(ap + 2, w * lo.z);
  atomicAdd(ap + 3, w * lo.w);
  atomicAdd(ap + 4, w * hi.x);
  atomicAdd(ap + 5, w * hi.y);
  atomicAdd(ap + 6, w * hi.z);
  atomicAdd(ap + 7, w * hi.w);
}

// ---------------- residual + bias + relu ----------------
__global__ void k_update(float* __restrict__ h, const float* __restrict__ agg,
                         const float* __restrict__ b, int total) {
  int t = blockIdx.x * blockDim.x + threadIdx.x;
  if (t < total) {
    float v = h[t] + agg[t] + b[t & 7];
    h[t] = v > 0.f ? v : 0.f;
  }
}

// ---------------- graph pooling ----------------
__global__ void k_pool(const float* __restrict__ h, const int* __restrict__ batch,
                       float* __restrict__ pooled, int total) {
  int t = blockIdx.x * blockDim.x + threadIdx.x;
  if (t < total) atomicAdd(&pooled[(size_t)batch[t >> 3] * CHN + (t & 7)], h[t]);
}

// ---------------- MLP head: relu(pooled @ hid_w + hid_b) @ out_w + out_b ----
// One wave handles 16 graphs; 8 hidden tiles of N=16, each = two fp32 WMMAs,
// hidden-dot with out_w fused and reduced across each 16-lane half.
__global__ __launch_bounds__(256) void k_mlp(
    const float* __restrict__ pooled, const float* __restrict__ hid_w,
    const float* __restrict__ hid_b, const float* __restrict__ out_w,
    const float* __restrict__ out_b, float* __restrict__ out, int g) {
  const int lane = threadIdx.x & 31;
  const int wave = blockIdx.x * (blockDim.x >> 5) + (threadIdx.x >> 5);
  const int base = wave * 16;
  if (base >= g) return;                         // wave-uniform
  const bool hiHalf = lane >= 16;
  const int m = lane & 15;
  const int row = min(base + m, g - 1);
  const int c0 = hiHalf ? 2 : 0;
  v2f A0, A1;
  A0[0] = pooled[(size_t)row * CHN + c0];
  A0[1] = pooled[(size_t)row * CHN + c0 + 1];
  A1[0] = pooled[(size_t)row * CHN + c0 + 4];
  A1[1] = pooled[(size_t)row * CHN + c0 + 5];

  float acc[8];
#pragma unroll
  for (int j = 0; j < 8; ++j) acc[j] = 0.f;

  const int k0 = hiHalf ? 2 : 0;
#pragma unroll
  for (int t = 0; t < HID / 16; ++t) {
    int col = t * 16 + m;
    v2f B0, B1;
    B0[0] = hid_w[(k0    ) * HID + col];
    B0[1] = hid_w[(k0 + 1) * HID + col];
    B1[0] = hid_w[(k0 + 4) * HID + col];
    B1[1] = hid_w[(k0 + 5) * HID + col];
    v8f d = {};
    d = __builtin_amdgcn_wmma_f32_16x16x4_f32(false, A0, false, B0, (short)0, d, false, false);
    d = __builtin_amdgcn_wmma_f32_16x16x4_f32(false, A1, false, B1, (short)0, d, false, false);
    float hb = hid_b[col];
    float ow = out_w[col];
#pragma unroll
    for (int j = 0; j < 8; ++j) {
      float v = d[j] + hb;
      acc[j] += (v > 0.f ? v : 0.f) * ow;        // graph (base+j+8*hiHalf), hidden col
    }
  }

  // Reduce hidden contributions across each 16-lane half.
#pragma unroll
  for (int j = 0; j < 8; ++j) {
    float s = acc[j];
    for (int msk = 1; msk < 16; msk <<= 1) s += __shfl_xor(s, msk, 32);
    int gi = base + j + (hiHalf ? 8 : 0);
    if (m == 0 && gi < g) out[gi] = s + out_b[0];
  }
}

// ---------------- host orchestration ----------------

extern "C" void kernel_launch(void* const* d_in, const int* in_sizes, int n_in,
                              void* d_out, int out_size, void* d_ws, size_t ws_size,
                              hipStream_t stream) {
  const int N = in_sizes[0];            // 500,000 nodes
  const int E = in_sizes[1] / 2;        // 8,000,000 edges
  const int G = out_size;               // 5,000 graphs

  const int*   x     = (const int*)d_in[0];
  const int*   e_src = (const int*)d_in[1];
  const int*   e_dst = e_src + E;
  const int*   batch = (const int*)d_in[2];
  const float* emb   = (const float*)d_in[3];
  const float* bn_g  = (const float*)d_in[4];
  const float* bn_b  = (const float*)d_in[5];
  const float* cw    = (const float*)d_in[6];
  const float* cb    = (const float*)d_in[7];
  const float* hidw  = (const float*)d_in[8];
  const float* hidb  = (const float*)d_in[9];
  const float* outw  = (const float*)d_in[10];
  const float* outb  = (const float*)d_in[11];
  float*       out   = (float*)d_out;

  // Workspace carve-out (~50 MB).
  char* p = (char*)d_ws;
  auto carve = [&](size_t bytes) {
    void* r = (void*)p;
    p += (bytes + 255) & ~(size_t)255;
    return r;
  };
  float* dinv   = (float*)carve((size_t)N * sizeof(float));
  float* h      = (float*)carve((size_t)N * CHN * sizeof(float));
  float* hw     = (float*)carve((size_t)N * CHN * sizeof(float));
  float* agg    = (float*)carve((size_t)N * CHN * sizeof(float));
  float* stats  = (float*)carve(64 * sizeof(float));
  float* pooled = (float*)carve((size_t)G * CHN * sizeof(float));

  const int totalNC = N * CHN;
  const dim3 B(256);
  auto blocks = [](int n) { return dim3((unsigned)((n + 255) / 256)); };

  // Degree (self loops contribute 1) -> dinv
  k_fill<<<blocks(N), B, 0, stream>>>(dinv, 1.0f, N);
  k_deg_acc<<<blocks(E), B, 0, stream>>>(e_dst, dinv, E);
  k_dinv<<<blocks(N), B, 0, stream>>>(dinv, N);

  // Embedding gather
  k_embed<<<blocks(totalNC), B, 0, stream>>>(x, emb, h, totalNC);

  const int tWaves  = (N + 15) / 16;
  const int tBlocks = (tWaves + 7) / 8;           // 8 waves per 256-thread block
  const int sTotal  = E + N;

  for (int i = 0; i < 8; ++i) {
    k_fill<<<1, 16, 0, stream>>>(stats, 0.f, 16);
    k_bn_stats<<<512, B, 0, stream>>>(h, stats, totalNC);
    k_transform<<<tBlocks, B, 0, stream>>>(h, stats, bn_g + i * CHN, bn_b + i * CHN,
                                           cw + i * CHN * CHN, hw, N);
    k_fill<<<blocks(totalNC), B, 0, stream>>>(agg, 0.f, totalNC);
    k_scatter<<<blocks(sTotal), B, 0, stream>>>(e_src, e_dst, dinv, hw, agg, E, N);
    k_update<<<blocks(totalNC), B, 0, stream>>>(h, agg, cb + i * CHN, totalNC);
  }

  // Pool + MLP head
  k_fill<<<blocks(G * CHN), B, 0, stream>>>(pooled, 0.f, G * CHN);
  k_pool<<<blocks(totalNC), B, 0, stream>>>(h, batch, pooled, totalNC);
  const int mWaves  = (G + 15) / 16;
  const int mBlocks = (mWaves + 7) / 8;
  k_mlp<<<mBlocks, B, 0, stream>>>(pooled, hidw, hidb, outw, outb, out, G);
}